// YOLOv1Loss_15152644621136
// MI455X (gfx1250) — compile-verified
//
#include <hip/hip_runtime.h>
#include <stdint.h>

// YOLOv1 loss for MI455X (gfx1250).
// Memory-bound streaming kernel: ~101.6 MB read -> one scalar (~4.4us HBM floor).
// CDNA5 Tensor Data Mover (tensor_load_to_lds + s_wait_tensorcnt) triple-buffers
// each sample's 5880B tile into LDS with depth-2 prefetch; 8 wave32 waves/WG.

#define S7       7
#define HW       49
#define NB       2        // boxes per cell
#define NC       20       // classes
#define NM       16       // targets per sample
#define CH       30       // 5*B + C channels
#define SAMPLE_F 1470     // HW*CH floats per sample
#define BLOCK    256      // 8 wave32 waves
#define SPB      8        // samples per workgroup (pipelined)
#define NBUF     3        // triple buffer -> depth-2 TDM prefetch

#if defined(__has_builtin)
# if __has_builtin(__builtin_amdgcn_tensor_load_to_lds)
#  define USE_TDM 1
# endif
#endif

typedef unsigned int u32x4 __attribute__((ext_vector_type(4)));
typedef int          i32x4 __attribute__((ext_vector_type(4)));
typedef int          i32x8 __attribute__((ext_vector_type(8)));

__device__ __forceinline__ float sigmoidf_(float x) {
  return 1.0f / (1.0f + __expf(-x));
}

__device__ __forceinline__ float iou_(float ax, float ay, float aw, float ah,
                                      float bx, float by, float bw, float bh) {
  float tlx = fmaxf(ax - aw * 0.5f, bx - bw * 0.5f);
  float tly = fmaxf(ay - ah * 0.5f, by - bh * 0.5f);
  float brx = fminf(ax + aw * 0.5f, bx + bw * 0.5f);
  float bry = fminf(ay + ah * 0.5f, by + bh * 0.5f);
  float inter = fmaxf(brx - tlx, 0.0f) * fmaxf(bry - tly, 0.0f);
  return inter / (aw * ah + bw * bh - inter);
}

#if defined(USE_TDM)
// Build a TDM descriptor (D#) per CDNA5 ISA and issue
// tensor_load_to_lds: one 1D tile of `ndw` dwords, global -> LDS.
__device__ __forceinline__ void tdm_load_f32(const float* gptr, uint32_t lds_byte,
                                             int ndw) {
  uint64_t ga = (uint64_t)(uintptr_t)gptr;
  // Group 0: [1:0]=count=1, [63:32]=lds_addr, [120:64]=global_addr, [127:126]=type 2
  u32x4 g0;
  g0.x = 1u;
  g0.y = lds_byte;
  g0.z = (uint32_t)ga;
  g0.w = ((uint32_t)(ga >> 32) & 0x01FFFFFFu) | (2u << 30);
  // Group 1: data_size=2 (4B), tensor_dim0=ndw, tensor_dim1=1, tile_dim0=ndw,
  // tile_dim1=1, tensor_dim0_stride=ndw, tensor_dim1_stride=ndw.
  uint32_t lo = (uint32_t)ndw & 0xFFFFu;
  uint32_t hi = (uint32_t)ndw >> 16;
  i32x8 g1;
  g1[0] = (int)(2u << 16);            // workgroup_mask=0, data_size=4B
  g1[1] = (int)(lo << 16);            // tensor_dim0[15:0] @ bits[63:48]
  g1[2] = (int)(hi | (1u << 16));     // tensor_dim0[31:16], tensor_dim1=1
  g1[3] = (int)(lo << 16);            // tile_dim0 @ bits[127:112]
  g1[4] = 1;                          // tile_dim1=1, tile_dim2=0
  g1[5] = ndw;                        // tensor_dim0_stride[31:0]
  g1[6] = (int)(lo << 16);            // stride0 hi=0, tensor_dim1_stride[15:0]
  g1[7] = (int)hi;                    // tensor_dim1_stride[47:16]
  i32x4 z4 = {0, 0, 0, 0};
# if __clang_major__ >= 23
  i32x8 z8 = {0, 0, 0, 0, 0, 0, 0, 0};
  __builtin_amdgcn_tensor_load_to_lds(g0, g1, z4, z4, z8, 0);
# else
  __builtin_amdgcn_tensor_load_to_lds(g0, g1, z4, z4, 0);
# endif
}
#endif

__global__ __launch_bounds__(BLOCK)
void yolo_loss_kernel(const float* __restrict__ outputs,
                      const float* __restrict__ targets,
                      float* __restrict__ out, int n) {
  __shared__ float  buf[NBUF][SAMPLE_F + 2]; // triple-buffered sample tiles
  __shared__ float  tg[NM * 5];              // staged targets of current sample
  __shared__ int    slot_m[HW * NB];         // target index owning (cell,box), else -1
  __shared__ int    cls_of_cell[HW];         // class id of target in cell, else -1
  __shared__ float  tgt_iou[NM];             // miou_best per target
  __shared__ float4 tgt_delta[NM];           // (dx, dy, w, h) per target
  __shared__ float  wave_sum[BLOCK / 32];

  const int tid = threadIdx.x;
  const int s0  = blockIdx.x * SPB;
  float acc = 0.0f;

#if defined(USE_TDM)
  const bool wave0 = (tid < 32);             // uniform per wave: only wave 0 issues TDM
  if (wave0) {                               // prime the pipeline: depth-2 prefetch
    if (s0 < n)
      tdm_load_f32(outputs + (size_t)s0 * SAMPLE_F,
                   (uint32_t)(uintptr_t)&buf[0][0], SAMPLE_F);
    if (SPB > 1 && s0 + 1 < n)
      tdm_load_f32(outputs + (size_t)(s0 + 1) * SAMPLE_F,
                   (uint32_t)(uintptr_t)&buf[1][0], SAMPLE_F);
  }
#endif

  for (int i = 0; i < SPB; ++i) {
    const int s = s0 + i;
    if (s >= n) break;
    const int bi = i % NBUF;
    const float* b = buf[bi];

#if defined(USE_TDM)
    // TDM ops of a wave complete in order: if load (s+1) is in flight, waiting
    // TENSORcnt<=1 guarantees load s has landed while keeping one tile in flight.
    const bool next_in_flight = (i + 1 < SPB) && (s + 1 < n);
    if (wave0) {
      if (next_in_flight) __builtin_amdgcn_s_wait_tensorcnt(1);
      else                __builtin_amdgcn_s_wait_tensorcnt(0);
    }
    __syncthreads();   // release all waves; also fences previous iteration's LDS use
    if (wave0 && (i + 2 < SPB) && (s + 2 < n))  // refill: buffer (i+2)%3 == (i-1)%3,
      tdm_load_f32(outputs + (size_t)(s + 2) * SAMPLE_F,   // free since iter i-1 done
                   (uint32_t)(uintptr_t)&buf[(i + 2) % NBUF][0], SAMPLE_F);
#else
    __syncthreads();
    for (int k = tid; k < SAMPLE_F; k += BLOCK)
      buf[bi][k] = outputs[(size_t)s * SAMPLE_F + k];
#endif

    // ---- clear scratch + stage targets (disjoint from in-flight TDM buffers) ----
    for (int k = tid; k < HW * NB; k += BLOCK) slot_m[k] = -1;
    for (int k = tid; k < HW; k += BLOCK)      cls_of_cell[k] = -1;
    for (int k = tid; k < NM * 5; k += BLOCK)  tg[k] = targets[(size_t)s * (NM * 5) + k];
    __syncthreads();

    // ---- per-target resolution: 16 threads, one target each ----
    if (tid < NM) {
      const float* g = &tg[tid * 5];
      float gx = g[0], gy = g[1], gw = g[2], gh = g[3];
      int   cls = (int)g[4];
      float fx = floorf(gx), fy = floorf(gy);
      int   cell = (int)fy * S7 + (int)fx;
      const float* cp = &b[cell * CH];
      float px[NB], py[NB], pw[NB], ph[NB];
#pragma unroll
      for (int q = 0; q < NB; ++q) {
        px[q] = sigmoidf_(cp[q * 4 + 0]) + fx;   // + column
        py[q] = sigmoidf_(cp[q * 4 + 1]) + fy;   // + row
        pw[q] = sigmoidf_(cp[q * 4 + 2]);
        ph[q] = sigmoidf_(cp[q * 4 + 3]);
      }
      float io0 = iou_(px[0], py[0], pw[0], ph[0], gx, gy, gw, gh);
      float io1 = iou_(px[1], py[1], pw[1], ph[1], gx, gy, gw, gh);
      int best = (io1 > io0) ? 1 : 0;            // argmax, first-wins on tie
      float bx = px[best], by = py[best], bw = pw[best], bh = ph[best];
      float mi = 0.0f;                           // IoUs are >= 0
#pragma unroll
      for (int mp = 0; mp < NM; ++mp) {
        const float* gp = &tg[mp * 5];
        mi = fmaxf(mi, iou_(bx, by, bw, bh, gp[0], gp[1], gp[2], gp[3]));
      }
      slot_m[cell * NB + best] = tid;
      cls_of_cell[cell] = cls;
      tgt_iou[tid] = mi;
      tgt_delta[tid] = make_float4(gx - fx, gy - fy, gw, gh);
    }
    __syncthreads();

    // ---- loss accumulation ----
    // conf + coord over 98 (cell,box) slots
    for (int idx = tid; idx < HW * NB; idx += BLOCK) {
      int hw = idx >> 1, bo = idx & 1;
      const float* cp = &b[hw * CH];
      float sc = sigmoidf_(cp[8 + bo]);
      int m = slot_m[idx];
      if (m >= 0) {
        float d0 = sc - tgt_iou[m];
        acc += d0 * d0;                          // LAM_OBJ = 1
        float4 d = tgt_delta[m];
        float sx = sigmoidf_(cp[bo * 4 + 0]);
        float sy = sigmoidf_(cp[bo * 4 + 1]);
        float sw = sigmoidf_(cp[bo * 4 + 2]);
        float sh = sigmoidf_(cp[bo * 4 + 3]);
        float ex = sx - d.x;
        float ey = sy - d.y;
        float ew = sqrtf(sw) - sqrtf(d.z);       // all args > 0 (sigmoid / wh>=0.2)
        float eh = sqrtf(sh) - sqrtf(d.w);
        acc += 5.0f * (ex * ex + ey * ey + ew * ew + eh * eh);  // LAM_COORD
      } else {
        acc += 0.25f * sc * sc;                  // (LAM_NOOBJ * sc)^2
      }
    }
    // class over 49*20 slots (only target cells contribute)
    for (int idx = tid; idx < HW * NC; idx += BLOCK) {
      int hw = idx / NC, c = idx - hw * NC;
      int cc = cls_of_cell[hw];
      if (cc >= 0) {
        float sp = sigmoidf_(b[hw * CH + 10 + c]);
        float t  = (c == cc) ? 1.0f : 0.0f;
        float e  = sp - t;
        acc += e * e;                            // LAM_CLASS = 1
      }
    }
    // loop-top barrier of next iteration separates scratch reuse
  }

  // ---- block reduction: wave32 shuffle, then cross-wave via LDS ----
  float v = acc;
#pragma unroll
  for (int o = 16; o > 0; o >>= 1) v += __shfl_down(v, o, 32);
  if ((tid & 31) == 0) wave_sum[tid >> 5] = v;
  __syncthreads();
  if (tid < (BLOCK / 32)) {
    float t = wave_sum[tid];
#pragma unroll
    for (int o = (BLOCK / 64); o > 0; o >>= 1) t += __shfl_down(t, o, BLOCK / 32);
    if (tid == 0) atomicAdd(out, t);
  }
}

__global__ void yolo_zero_kernel(float* __restrict__ out, int nn) {
  int i = blockIdx.x * blockDim.x + threadIdx.x;
  if (i < nn) out[i] = 0.0f;
}

extern "C" void kernel_launch(void* const* d_in, const int* in_sizes, int n_in,
                              void* d_out, int out_size, void* d_ws, size_t ws_size,
                              hipStream_t stream) {
  const float* outputs = (const float*)d_in[0];
  const float* targets = (const float*)d_in[1];
  float* out = (float*)d_out;
  const int n = in_sizes[0] / SAMPLE_F;

  yolo_zero_kernel<<<(out_size + 255) / 256, 256, 0, stream>>>(out, out_size);
  const int blocks = (n + SPB - 1) / SPB;
  yolo_loss_kernel<<<blocks, BLOCK, 0, stream>>>(outputs, targets, out, n);
}